// VectorQuantizer_42339787604548
// MI455X (gfx1250) — compile-verified
//
#include <hip/hip_runtime.h>
#include <hip/hip_bf16.h>

// ---------------------------------------------------------------------------
// VQ quantizer for MI455X (gfx1250): f32 WMMA GEMM for the distance matrix,
// fused argmin epilogue, async global->LDS double buffering (ASYNCcnt).
// ---------------------------------------------------------------------------

typedef float v2f __attribute__((ext_vector_type(2)));
typedef float v8f __attribute__((ext_vector_type(8)));
typedef int   vq_i32x4 __attribute__((vector_size(16)));

#define VQ_B     4
#define VQ_C     256
#define VQ_S     8192            // T*H*W = 8*32*32
#define VQ_N     32768           // B*S
#define VQ_K     1024            // codebook size
#define VQ_ELEMS 8388608         // B*C*S

// d_out layout (floats): [zq 8388608][vq_loss 1][inds 32768][d 32768*1024]
#define OUT_ZQ   0
#define OUT_LOSS 8388608
#define OUT_IND  8388609
#define OUT_D    8421377

// d_ws layout (float offsets)
#define WS_FLAT   0              // 8,388,608 floats: flat [N][C]
#define WS_XN     8388608        // 32768 floats
#define WS_EN     8421376        // 1024 floats
#define WS_INDI   8422400        // 32768 int32
#define WS_ACC    8455168        // 1 float accumulator
#define WS_ROWMIN 8455172        // 32768 u64 (even float offset -> 8B aligned)

// Async global->LDS path (gfx1250); falls back to sync staging if unavailable.
#if defined(__HIP_DEVICE_COMPILE__) && defined(__gfx1250__) &&                 \
    __has_builtin(__builtin_amdgcn_global_load_async_to_lds_b128) &&           \
    __has_builtin(__builtin_amdgcn_global_load_async_to_lds_b32) &&            \
    __has_builtin(__builtin_amdgcn_s_wait_asynccnt)
#define VQ_ASYNC_LDS 1
// Builtin params are typed, non-const, AS-qualified pointers:
//   b128: (int4 AS1* src, int4 AS3* dst, Ii offset, Ii cpol)
//   b32 : (int  AS1* src, int  AS3* dst, Ii offset, Ii cpol)
#define VQ_GLB4(p) ((__attribute__((address_space(1))) vq_i32x4*)(p))
#define VQ_LDS4(p) ((__attribute__((address_space(3))) vq_i32x4*)(p))
#define VQ_GLB1(p) ((__attribute__((address_space(1))) int*)(p))
#define VQ_LDS1(p) ((__attribute__((address_space(3))) int*)(p))
#else
#define VQ_ASYNC_LDS 0
#endif

#define APITCH 36                // A-tile pitch (16B-aligned rows, no bank conflicts)
#define EPITCH 132               // Et pitch

__device__ __forceinline__ unsigned int vq_fkey(float f) {
    unsigned int u = __float_as_uint(f);
    return (u & 0x80000000u) ? ~u : (u | 0x80000000u);
}

// ---------------------------------------------------------------------------
// K0: zero accumulator + xnorm, seed per-row argmin slots
__global__ __launch_bounds__(256) void vq_init_kernel(
    float* __restrict__ acc, float* __restrict__ xnorm,
    unsigned long long* __restrict__ rowmin)
{
    const int i = blockIdx.x * 256 + threadIdx.x;
    xnorm[i]  = 0.0f;
    rowmin[i] = 0xFFFFFFFFFFFFFFFFull;
    if (i == 0) acc[0] = 0.0f;
}

// ---------------------------------------------------------------------------
// K1: transpose z (B,C,S) -> flat (B*S, C), LDS-tiled; fused ||x||^2 partials
__global__ __launch_bounds__(256) void vq_transpose_kernel(
    const float* __restrict__ z, float* __restrict__ flat, float* __restrict__ xnorm)
{
    __shared__ float tile[32][33];
    const int sBase = blockIdx.x * 32;
    const int cBase = blockIdx.y * 32;
    const int b     = blockIdx.z;
    const int tx = threadIdx.x;      // 0..31 (one wave per ty row)
    const int ty = threadIdx.y;      // 0..7
#pragma unroll
    for (int j = 0; j < 32; j += 8)
        tile[ty + j][tx] = z[((size_t)(b * VQ_C + cBase + ty + j)) * VQ_S + sBase + tx];
    __syncthreads();
#pragma unroll
    for (int j = 0; j < 32; j += 8) {
        const int row = b * VQ_S + sBase + ty + j;
        const float v = tile[tx][ty + j];
        flat[(size_t)row * VQ_C + cBase + tx] = v;
        float sq = v * v;
#pragma unroll
        for (int off = 16; off; off >>= 1) sq += __shfl_xor(sq, off, 32);
        if (tx == 0) atomicAdd(&xnorm[row], sq);
    }
}

// ---------------------------------------------------------------------------
// K2: squared row norms of E [1024][256] (wave32 per row)
__global__ __launch_bounds__(256) void vq_rownorm_kernel(
    const float* __restrict__ src, float* __restrict__ dst, int rows)
{
    const int w    = threadIdx.x >> 5;
    const int lane = threadIdx.x & 31;
    const int row  = blockIdx.x * 8 + w;
    if (row >= rows) return;
    const float* p = src + (size_t)row * VQ_C;
    float s = 0.0f;
#pragma unroll
    for (int j = lane; j < VQ_C; j += 32) { float v = p[j]; s += v * v; }
#pragma unroll
    for (int off = 16; off; off >>= 1) s += __shfl_xor(s, off, 32);
    if (lane == 0) dst[row] = s;
}

// ---------------------------------------------------------------------------
// K3: d[n][k] = xn[n] + en[k] - 2 * flat[n].E[k]   via V_WMMA_F32_16X16X4_F32
// Workgroup: 256 threads (8 waves), 128(M) x 128(Kcols) output tile.
// Async path: double-buffered LDS staging via GLOBAL_LOAD_ASYNC_TO_LDS.
// Epilogue writes d AND reduces per-row argmin into rowmin via atomic u64 min.
__global__ __launch_bounds__(256) void vq_gemm_kernel(
    const float* __restrict__ flat, const float* __restrict__ E,
    const float* __restrict__ xnorm, const float* __restrict__ enorm,
    float* __restrict__ dmat, unsigned long long* __restrict__ rowmin)
{
#if VQ_ASYNC_LDS
    __shared__ float Atile[2][128][APITCH];
    __shared__ float Et[2][32][EPITCH];
#else
    __shared__ float Atile[1][128][APITCH];
    __shared__ float Et[1][32][EPITCH];
#endif
    __shared__ float xn_s[128];
    __shared__ float en_s[128];

    const int rowBase = blockIdx.x * 128;
    const int colBase = blockIdx.y * 128;
    const int tid  = threadIdx.x;
    const int lane = tid & 31;
    const int w    = tid >> 5;
    const int wm   = w & 3;
    const int wn   = w >> 2;

    if (tid < 128) {
        xn_s[tid] = xnorm[rowBase + tid];
        en_s[tid] = enorm[colBase + tid];
    }

    v8f acc[2][4];
    {
        v8f zero = {0.f, 0.f, 0.f, 0.f, 0.f, 0.f, 0.f, 0.f};
#pragma unroll
        for (int mt = 0; mt < 2; ++mt)
#pragma unroll
            for (int nt = 0; nt < 4; ++nt) acc[mt][nt] = zero;
    }

    const int m16 = lane & 15;
    const int kb  = (lane >> 4) << 1;   // lanes 0-15: K=0,1 ; lanes 16-31: K=2,3
    const int r_st   = tid >> 1;        // staging row / col
    const int cb4_st = (tid & 1) * 16;  // staging c-offset

#if VQ_ASYNC_LDS
    // Issue chunk 0 into buffer 0.
    {
        const float* ga = flat + (size_t)(rowBase + r_st) * VQ_C + cb4_st;
        float* la = &Atile[0][r_st][cb4_st];
        __builtin_amdgcn_global_load_async_to_lds_b128(VQ_GLB4(ga), VQ_LDS4(la), 0, 0);
        __builtin_amdgcn_global_load_async_to_lds_b128(VQ_GLB4(ga), VQ_LDS4(la), 16, 0);
        __builtin_amdgcn_global_load_async_to_lds_b128(VQ_GLB4(ga), VQ_LDS4(la), 32, 0);
        __builtin_amdgcn_global_load_async_to_lds_b128(VQ_GLB4(ga), VQ_LDS4(la), 48, 0);
        const float* ge = E + (size_t)(colBase + r_st) * VQ_C + cb4_st;
#pragma unroll
        for (int i = 0; i < 16; ++i)
            __builtin_amdgcn_global_load_async_to_lds_b32(
                VQ_GLB1(ge + i), VQ_LDS1(&Et[0][cb4_st + i][r_st]), 0, 0);
    }
#endif

    for (int ci = 0; ci < 8; ++ci) {
        const int cb = ci * 32;
#if VQ_ASYNC_LDS
        const int buf = ci & 1;
        __builtin_amdgcn_s_wait_asynccnt(0);   // chunk ci landed in LDS
        __syncthreads();                       // all waves done reading buf^1
        if (ci + 1 < 8) {
            const int nb = buf ^ 1;
            const float* ga = flat + (size_t)(rowBase + r_st) * VQ_C + cb + 32 + cb4_st;
            float* la = &Atile[nb][r_st][cb4_st];
            __builtin_amdgcn_global_load_async_to_lds_b128(VQ_GLB4(ga), VQ_LDS4(la), 0, 0);
            __builtin_amdgcn_global_load_async_to_lds_b128(VQ_GLB4(ga), VQ_LDS4(la), 16, 0);
            __builtin_amdgcn_global_load_async_to_lds_b128(VQ_GLB4(ga), VQ_LDS4(la), 32, 0);
            __builtin_amdgcn_global_load_async_to_lds_b128(VQ_GLB4(ga), VQ_LDS4(la), 48, 0);
            const float* ge = E + (size_t)(colBase + r_st) * VQ_C + cb + 32 + cb4_st;
#pragma unroll
            for (int i = 0; i < 16; ++i)
                __builtin_amdgcn_global_load_async_to_lds_b32(
                    VQ_GLB1(ge + i), VQ_LDS1(&Et[nb][cb4_st + i][r_st]), 0, 0);
        }
#else
        const int buf = 0;
        __syncthreads();
        {
            const float4* src = (const float4*)(flat + (size_t)(rowBase + r_st) * VQ_C + cb + cb4_st);
#pragma unroll
            for (int i = 0; i < 4; ++i) {
                float4 v = src[i];
                Atile[0][r_st][cb4_st + 4 * i + 0] = v.x;
                Atile[0][r_st][cb4_st + 4 * i + 1] = v.y;
                Atile[0][r_st][cb4_st + 4 * i + 2] = v.z;
                Atile[0][r_st][cb4_st + 4 * i + 3] = v.w;
            }
            if (cb + 32 < VQ_C)
                __builtin_prefetch(flat + (size_t)(rowBase + r_st) * VQ_C + cb + 32 + cb4_st, 0, 0);
        }
        {
            const float4* src = (const float4*)(E + (size_t)(colBase + r_st) * VQ_C + cb + cb4_st);
#pragma unroll
            for (int i = 0; i < 4; ++i) {
                float4 v = src[i];
                Et[0][cb4_st + 4 * i + 0][r_st] = v.x;
                Et[0][cb4_st + 4 * i + 1][r_st] = v.y;
                Et[0][cb4_st + 4 * i + 2][r_st] = v.z;
                Et[0][cb4_st + 4 * i + 3][r_st] = v.w;
            }
        }
        __syncthreads();
#endif

#pragma unroll
        for (int ks = 0; ks < 8; ++ks) {
            const int c0 = ks * 4;
            v2f a[2], b[4];
#pragma unroll
            for (int mt = 0; mt < 2; ++mt) {
                const int r = wm * 32 + mt * 16 + m16;
                a[mt].x = Atile[buf][r][c0 + kb];
                a[mt].y = Atile[buf][r][c0 + kb + 1];
            }
#pragma unroll
            for (int nt = 0; nt < 4; ++nt) {
                const int cC = wn * 64 + nt * 16 + m16;
                b[nt].x = Et[buf][c0 + kb][cC];
                b[nt].y = Et[buf][c0 + kb + 1][cC];
            }
#pragma unroll
            for (int mt = 0; mt < 2; ++mt)
#pragma unroll
                for (int nt = 0; nt < 4; ++nt)
                    acc[mt][nt] = __builtin_amdgcn_wmma_f32_16x16x4_f32(
                        false, a[mt], false, b[nt], (short)0, acc[mt][nt], false, false);
        }
    }

    // Epilogue. C/D layout: VGPR vi -> lanes 0-15: M=vi, N=lane; lanes 16-31: M=vi+8.
    // For each (mt, vi): write d for 4 col-tiles, pack (key|col), butterfly-min
    // across the 16 lanes of the half-wave (xor 1,2,4,8 stays in-half), atomicMin.
    const int mhi = (lane >> 4) << 3;
#pragma unroll
    for (int mt = 0; mt < 2; ++mt) {
#pragma unroll
        for (int vi = 0; vi < 8; ++vi) {
            const int rowL = wm * 32 + mt * 16 + vi + mhi;
            const float xn = xn_s[rowL];
            unsigned long long best = 0xFFFFFFFFFFFFFFFFull;
#pragma unroll
            for (int nt = 0; nt < 4; ++nt) {
                const int colL = wn * 64 + nt * 16 + m16;
                const int col  = colBase + colL;
                const float dv = xn + en_s[colL] - 2.0f * acc[mt][nt][vi];
                dmat[(size_t)(rowBase + rowL) * VQ_K + col] = dv;
                const unsigned long long p =
                    ((unsigned long long)vq_fkey(dv) << 32) | (unsigned int)col;
                if (p < best) best = p;
            }
#pragma unroll
            for (int off = 1; off < 16; off <<= 1) {
                unsigned long long o = __shfl_xor(best, off, 32);
                if (o < best) best = o;
            }
            if (m16 == 0)
                atomicMin(&rowmin[rowBase + rowL], best);
        }
    }
}

// ---------------------------------------------------------------------------
// K4: decode packed argmin -> int + float index outputs
__global__ __launch_bounds__(256) void vq_decode_kernel(
    const unsigned long long* __restrict__ rowmin,
    int* __restrict__ inds_i, float* __restrict__ inds_f)
{
    const int row = blockIdx.x * 256 + threadIdx.x;
    const int k = (int)(rowmin[row] & 0xFFFFFFFFull);
    inds_i[row] = k;
    inds_f[row] = (float)k;
}

// ---------------------------------------------------------------------------
// K5: z_q gather (into B,C,T,H,W layout) + fused sum((zq-z)^2)
__global__ __launch_bounds__(256) void vq_gather_loss_kernel(
    const float* __restrict__ z, const float* __restrict__ E,
    const int* __restrict__ inds, float* __restrict__ zq_out, float* __restrict__ acc)
{
    __shared__ float wsum[8];
    const size_t idx = (size_t)blockIdx.x * 256 + threadIdx.x;
    const size_t CS  = (size_t)VQ_C * VQ_S;
    const int b = (int)(idx / CS);
    const size_t rem = idx % CS;
    const int c = (int)(rem / VQ_S);
    const int s = (int)(rem % VQ_S);
    const int k = inds[b * VQ_S + s];
    const float zq = E[(size_t)k * VQ_C + c];
    const float zv = z[idx];
    zq_out[idx] = zq;                     // z_q_ste == z_q numerically
    float d = zq - zv;
    float sq = d * d;
#pragma unroll
    for (int off = 16; off; off >>= 1) sq += __shfl_xor(sq, off, 32);
    const int lane = threadIdx.x & 31, wv = threadIdx.x >> 5;
    if (lane == 0) wsum[wv] = sq;
    __syncthreads();
    if (threadIdx.x == 0) {
        float t = 0.0f;
#pragma unroll
        for (int i = 0; i < 8; ++i) t += wsum[i];
        atomicAdd(acc, t);
    }
}

// ---------------------------------------------------------------------------
// K6: vq_loss = (1 + 0.25) * mean((zq - z)^2)
__global__ void vq_finalize_kernel(const float* __restrict__ acc, float* __restrict__ out_loss)
{
    out_loss[0] = 1.25f * acc[0] / (float)VQ_ELEMS;
}

// ---------------------------------------------------------------------------
extern "C" void kernel_launch(void* const* d_in, const int* in_sizes, int n_in,
                              void* d_out, int out_size, void* d_ws, size_t ws_size,
                              hipStream_t stream) {
    const float* z = (const float*)d_in[0];
    const float* E = (const float*)d_in[1];
    float* out = (float*)d_out;
    float* ws  = (float*)d_ws;

    float* flat   = ws + WS_FLAT;
    float* xnorm  = ws + WS_XN;
    float* enorm  = ws + WS_EN;
    int*   inds_i = (int*)(ws + WS_INDI);
    float* acc    = ws + WS_ACC;
    unsigned long long* rowmin = (unsigned long long*)(ws + WS_ROWMIN);

    float* zq_out  = out + OUT_ZQ;
    float* loss_o  = out + OUT_LOSS;
    float* inds_f  = out + OUT_IND;
    float* dmat    = out + OUT_D;

    vq_init_kernel<<<VQ_N / 256, 256, 0, stream>>>(acc, xnorm, rowmin);

    dim3 tgrid(VQ_S / 32, VQ_C / 32, VQ_B);
    vq_transpose_kernel<<<tgrid, dim3(32, 8), 0, stream>>>(z, flat, xnorm);

    vq_rownorm_kernel<<<VQ_K / 8, 256, 0, stream>>>(E, enorm, VQ_K);

    dim3 ggrid(VQ_N / 128, VQ_K / 128);
    vq_gemm_kernel<<<ggrid, 256, 0, stream>>>(flat, E, xnorm, enorm, dmat, rowmin);

    vq_decode_kernel<<<VQ_N / 256, 256, 0, stream>>>(rowmin, inds_i, inds_f);

    vq_gather_loss_kernel<<<VQ_ELEMS / 256, 256, 0, stream>>>(z, E, inds_i, zq_out, acc);

    vq_finalize_kernel<<<1, 1, 0, stream>>>(acc, loss_o);
}